// relational_model_31507880083779
// MI455X (gfx1250) — compile-verified
//
#include <hip/hip_runtime.h>
#include <cstddef>

#define BATCH 64
#define SEQ   2048
#define DIM   256
#define NREL  3
#define MTOT  128
#define MC    127          // MTOT - 1
#define S1    260          // padded LDS row stride (floats): bank = (4*row + col) % 64 -> conflict free

typedef __attribute__((ext_vector_type(2))) float v2f;
typedef __attribute__((ext_vector_type(8))) float v8f;

// ---------------------------------------------------------------------------
// Kernel 1: per-batch compaction of mask -> ascending indices of ones + count.
// One wave32 per batch; ballot scan, lane 0 extracts set bits (total work ~= #ones).
// ---------------------------------------------------------------------------
__global__ __launch_bounds__(32) void mask_scan_kernel(const int* __restrict__ mask,
                                                       int* __restrict__ idx,
                                                       int* __restrict__ ncomp) {
    const int b    = blockIdx.x;
    const int lane = threadIdx.x;
    const int* m   = mask + b * SEQ;
    int cnt = 0;
    for (int base = 0; base < SEQ; base += 32) {
        int v = m[base + lane];
        unsigned long long bal = __ballot(v != 0);
        if (lane == 0) {
            unsigned bm = (unsigned)bal;          // wave32: low 32 bits
            while (bm && cnt < MC) {
                int bit = __ffs(bm) - 1;
                idx[b * MC + cnt] = base + bit;
                ++cnt;
                bm &= bm - 1;
            }
        }
    }
    if (lane == 0) ncomp[b] = cnt;                // n = min(sum(mask), 127)
}

// ---------------------------------------------------------------------------
// Kernel 1b: one-time transpose Wt[r][d][k] = W[k][3d + r]  (WMMA-B-friendly
// layout: per lane (N=d) the K values are contiguous -> single b64 frag loads).
// ---------------------------------------------------------------------------
__global__ __launch_bounds__(256) void w_transpose_kernel(const float* __restrict__ W,
                                                          float* __restrict__ Wt) {
    int t = blockIdx.x * 256 + threadIdx.x;       // t in [0, 3*256*256)
    if (t >= NREL * DIM * DIM) return;
    const int r   = t / (DIM * DIM);
    const int rem = t - r * (DIM * DIM);
    const int d   = rem / DIM;
    const int k   = rem - d * DIM;
    Wt[t] = W[k * (NREL * DIM) + 3 * d + r];
}

// ---------------------------------------------------------------------------
// Kernel 2: one block per (batch, relation). 8 waves, static LDS (compile-checked).
//   sF = from-matrix [128 x 256] f32 (row 0 = ones, rows 1..n gathered, rest 0)
//   GEMM1: sT = sF @ Wr  via V_WMMA_F32_16X16X4_F32 (B from Wt if available)
//   GEMM2: E  = sF @ sT^T, fused -inf masking, write [B,128,128,3] f32
// ---------------------------------------------------------------------------
template <bool USE_WT>
__global__ __launch_bounds__(256) void relational_kernel(const float* __restrict__ embds, // [B,S,D]
                                                         const float* __restrict__ Wsrc,  // W or Wt
                                                         const int*   __restrict__ idx,   // [B,MC]
                                                         const int*   __restrict__ ncomp, // [B]
                                                         float*       __restrict__ out)   // [B,M,M,3]
{
    __shared__ float sF[MTOT * S1];   // 133,120 B
    __shared__ float sT[MTOT * S1];   // 133,120 B  (266,240 B total < 320 KB WGP LDS)

    const int b    = blockIdx.x;
    const int r    = blockIdx.y;
    const int tid  = threadIdx.x;
    const int wave = tid >> 5;
    const int lane = tid & 31;
    const int hl   = lane >> 4;     // which 16-lane half
    const int lm   = lane & 15;

    const int  n    = ncomp[b];
    const int* bidx = idx + b * MC;
    const float* e  = embds + (size_t)b * SEQ * DIM;
    const float NEG_INF = -__builtin_huge_valf();

    // ---- Phase 1: stage from-matrix into LDS (wave w handles rows w, w+8, ...) ----
    for (int row = wave; row < MTOT; row += 8) {
        float* dst = sF + row * S1;
        if (row == 0) {
            for (int c = lane; c < DIM; c += 32) dst[c] = 1.0f;
        } else if (row - 1 < n) {
            const float* src = e + (size_t)bidx[row - 1] * DIM;
            for (int c = lane * 4; c < DIM; c += 128) {
                float4 v = *(const float4*)(src + c);
                *(float4*)(dst + c) = v;
            }
        } else {
            for (int c = lane; c < DIM; c += 32) dst[c] = 0.0f;
        }
    }
    __syncthreads();

    // ---- GEMM1: T[j,d] = sum_k F[j,k] * W[k, 3d+r]; wave w owns T rows [16w,16w+16) ----
    {
        const int jr = wave * 16;
        const float* fRow = sF + (jr + lm) * S1;        // A: M = lane%16, K in VGPR slots
        for (int dt = 0; dt < 16; ++dt) {
            const int dn = dt * 16 + lm;                // B: N = lane%16
            const float* wRow = USE_WT ? (Wsrc + ((size_t)r * DIM + dn) * DIM) : nullptr;
            v8f acc = {};
            for (int k0 = 0; k0 < DIM; k0 += 4) {
                const int ka = k0 + 2 * hl;             // lanes 16-31 hold K+2, K+3
                v2f a = *(const v2f*)(fRow + ka);       // contiguous 8B LDS load
                v2f bb;
                if (USE_WT) {
                    bb = *(const v2f*)(wRow + ka);      // contiguous 8B global load
                } else {
                    bb.x = Wsrc[ka * (NREL * DIM) + 3 * dn + r];
                    bb.y = Wsrc[(ka + 1) * (NREL * DIM) + 3 * dn + r];
                }
                acc = __builtin_amdgcn_wmma_f32_16x16x4_f32(
                        false, a, false, bb, (short)0, acc, false, false);
            }
            // D layout: VGPR v, lane -> M = v + 8*hl, N = lane%16
            float* tBase = sT + (jr + 8 * hl) * S1 + dt * 16 + lm;
#pragma unroll
            for (int v = 0; v < 8; ++v) tBase[v * S1] = acc[v];
        }
    }
    __syncthreads();

    // ---- GEMM2: E[i,j] = sum_d F[i,d] * T[j,d]; wave w owns i-tile w ----
    {
        const int ir = wave * 16;
        const float* fRow = sF + (ir + lm) * S1;
        for (int jt = 0; jt < 8; ++jt) {
            const float* tRow = sT + (jt * 16 + lm) * S1;  // B[d, j] = T[j, d]
            v8f acc = {};
            for (int d0 = 0; d0 < DIM; d0 += 4) {
                const int da = d0 + 2 * hl;
                v2f a  = *(const v2f*)(fRow + da);
                v2f bb = *(const v2f*)(tRow + da);
                acc = __builtin_amdgcn_wmma_f32_16x16x4_f32(
                        false, a, false, bb, (short)0, acc, false, false);
            }
#pragma unroll
            for (int v = 0; v < 8; ++v) {
                const int i = ir + v + 8 * hl;
                const int j = jt * 16 + lm;
                const bool iv = (i >= 1) && (i <= n);          // row 0 and padded rows -> -inf
                const bool jv = (j == 0) || (j <= n);          // col 0 valid; padded cols -> -inf
                const float val = (iv && jv) ? acc[v] : NEG_INF;
                out[(((size_t)b * MTOT + i) * MTOT + j) * NREL + r] = val;
            }
        }
    }
}

// ---------------------------------------------------------------------------
extern "C" void kernel_launch(void* const* d_in, const int* in_sizes, int n_in,
                              void* d_out, int out_size, void* d_ws, size_t ws_size,
                              hipStream_t stream) {
    const float* embds = (const float*)d_in[0];
    const int*   mask  = (const int*)d_in[1];
    const float* W     = (const float*)d_in[2];
    float*       out   = (float*)d_out;

    // ws layout: [0, 32KB) = idx + ncomp; [32KB, 32KB + 768KB) = Wt (optional)
    int*   idx    = (int*)d_ws;                            // [BATCH][MC]
    int*   ncomp  = idx + BATCH * MC;                      // [BATCH]  (ends exactly at 32 KB)
    float* Wt     = (float*)((char*)d_ws + 32768);
    const size_t wtBytes = (size_t)NREL * DIM * DIM * sizeof(float);  // 786,432 B
    const bool   useWt   = ws_size >= (size_t)32768 + wtBytes;        // deterministic per run

    mask_scan_kernel<<<BATCH, 32, 0, stream>>>(mask, idx, ncomp);

    if (useWt) {
        const int nEl = NREL * DIM * DIM;
        w_transpose_kernel<<<(nEl + 255) / 256, 256, 0, stream>>>(W, Wt);
        relational_kernel<true><<<dim3(BATCH, NREL), 256, 0, stream>>>(embds, Wt, idx, ncomp, out);
    } else {
        relational_kernel<false><<<dim3(BATCH, NREL), 256, 0, stream>>>(embds, W, idx, ncomp, out);
    }
}